// ACT_54838142435966
// MI455X (gfx1250) — compile-verified
//
#include <hip/hip_runtime.h>
#include <math.h>

// Problem constants (from reference setup_inputs)
#define S_DIM 128
#define B_DIM 512
#define I_DIM 512
#define H_DIM 256
#define O_DIM 256
#define MAX_N 10
// Padded LDS row strides (floats). stride % 64 == 4 => A-frag float2 loads
// (lane L: addr = m*stride + k0 + 2*(L/16)) hit 32 distinct banks.
#define SROW 260   // hidden-state buffers (rows of 256)
#define XROW 516   // staged x tile (rows of 512)

typedef __attribute__((ext_vector_type(2))) float v2f;
typedef __attribute__((ext_vector_type(4))) float v4f;
typedef __attribute__((ext_vector_type(8))) float v8f;

// D = A(16x4 f32) * B(4x16 f32) + C(16x16 f32)  ->  v_wmma_f32_16x16x4_f32
// A frag: lane L (m = L%16, khi = L/16): v0 = A[m][2*khi],   v1 = A[m][2*khi+1]
// B frag: lane L (n = L%16, khi = L/16): v0 = B[2*khi][n],   v1 = B[2*khi+1][n]
// C/D:    lane L reg r -> (M = 8*(L/16)+r, N = L%16)
static __device__ inline v8f wmma_f32_k4(v2f a, v2f b, v8f c) {
  return __builtin_amdgcn_wmma_f32_16x16x4_f32(false, a, false, b,
                                               (short)0, c, false, false);
}

static __device__ inline v8f v8_splat(float v) {
  v8f r;
#pragma unroll
  for (int i = 0; i < 8; ++i) r[i] = v;
  return r;
}

__global__ __launch_bounds__(512) void act_persistent_kernel(
    const float* __restrict__ x,       // [S,B,I]
    const float* __restrict__ W_ih,    // [H, I+1] row-major
    const float* __restrict__ b_ih,    // [H]
    const float* __restrict__ W_hh,    // [H, H]
    const float* __restrict__ b_hh,    // [H]
    const float* __restrict__ W_halt,  // [1, H]
    const float* __restrict__ b_halt,  // [1]
    const float* __restrict__ W_out,   // [O, H]
    const float* __restrict__ b_out,   // [O]
    float* __restrict__ y_out,         // [S,B,O]
    float* __restrict__ p_out,         // [S,B]
    float* __restrict__ n_out)         // [S,B]
{
  // Double-buffered hidden state + staged x tile for this WG's 16 batch rows.
  __shared__ __align__(16) float s_ls[2][16 * SROW];
  __shared__ __align__(16) float xs[16 * XROW];
  __shared__ float whalt_s[H_DIM];
  __shared__ float partial[16][33];     // halt matvec partials (padded)
  __shared__ float p_s[16];             // p_n per batch row
  __shared__ float hsum_s[16], rem_s[16], nst_s[16], run_s[16];

  const int tid  = threadIdx.x;
  const int wave = tid >> 5;            // 0..15 : one 16-col N-tile per wave
  const int lane = tid & 31;
  const int ln   = lane & 15;           // M row (A/C) or N col (B/C)
  const int khi  = lane >> 4;           // K-pair / M-half selector
  const int b0   = blockIdx.x * 16;     // batch tile base (grid = 32)
  const int col  = wave * 16 + ln;      // this wave's column (H and O spaces)
  const int m    = ln;                  // A-frag row for this lane

  // Per-lane column constants (uniform across the 8 C regs of a lane).
  const float bsum = b_ih[col] + b_hh[col];
  const float wf   = W_ih[(size_t)col * (I_DIM + 1) + I_DIM];  // flag column
  const float bo   = b_out[col];
  const float bh   = b_halt[0];

  if (tid < H_DIM) whalt_s[tid] = W_halt[tid];
  for (int i = tid; i < 2 * 16 * SROW; i += 512) s_ls[0][i] = 0.0f;
  if (tid < 16) {
    hsum_s[tid] = 0.0f; rem_s[tid] = 0.0f; nst_s[tid] = 0.0f;
    run_s[tid] = 1.0f;  p_s[tid] = 0.0f;
  }
  __syncthreads();

  int cur = 0;

  for (int t = 0; t < S_DIM; ++t) {
    // ---- stage x tile [16 x 512] into LDS (coalesced float4) ----
    {
      const float* xsrc = x + ((size_t)t * B_DIM + b0) * I_DIM;
#pragma unroll
      for (int i = 0; i < 4; ++i) {
        const int idx = tid + i * 512;        // 2048 float4 chunks total
        const int r   = idx >> 7;             // 128 float4 per row
        const int c   = (idx & 127) << 2;
        *(v4f*)(xs + r * XROW + c) = *(const v4f*)(xsrc + (size_t)r * I_DIM + c);
      }
    }
    // prefetch next timestep's x tile; the 10 ponder steps hide the HBM fetch
    if (t + 1 < S_DIM) {
      const char* nx = (const char*)(x + ((size_t)(t + 1) * B_DIM + b0) * I_DIM);
      __builtin_prefetch(nx + (size_t)tid * 64, 0, 1);  // 512 thr * 64B = 32KB
    }
    __syncthreads();

    // ---- base = x_t @ W_ih[:, :I].T + (b_ih + b_hh); two K-split chains ----
    v8f baseA = v8_splat(bsum);
    v8f baseB = v8_splat(0.0f);
    {
      const float* wr = W_ih + (size_t)col * (I_DIM + 1);
#pragma unroll 8
      for (int k0 = 0; k0 < I_DIM / 2; k0 += 4) {
        const int kA = k0 + 2 * khi;
        const int kB = kA + I_DIM / 2;
        v2f aA = *(const v2f*)(xs + m * XROW + kA);
        v2f aB = *(const v2f*)(xs + m * XROW + kB);
        v2f bA; bA[0] = wr[kA]; bA[1] = wr[kA + 1];  // odd stride: scalar loads
        v2f bB; bB[0] = wr[kB]; bB[1] = wr[kB + 1];
        baseA = wmma_f32_k4(aA, bA, baseA);
        baseB = wmma_f32_k4(aB, bB, baseB);
      }
    }
    v8f base;
#pragma unroll
    for (int r = 0; r < 8; ++r) base[r] = baseA[r] + baseB[r];

    v8f yacc = v8_splat(0.0f);
    v8f sacc = v8_splat(0.0f);

    for (int n = 0; n < MAX_N; ++n) {
      // ---- pre = base (+flag col if n==0) + s_cur @ W_hh.T ; s_new = tanh ----
      const float flagv = (n == 0) ? wf : 0.0f;
      v8f accA, accB = v8_splat(0.0f);
#pragma unroll
      for (int r = 0; r < 8; ++r) accA[r] = base[r] + flagv;
      {
        const float* scur = &s_ls[cur][0];
        const float* wh   = W_hh + (size_t)col * H_DIM;
#pragma unroll 8
        for (int k0 = 0; k0 < H_DIM / 2; k0 += 4) {
          const int kA = k0 + 2 * khi;
          const int kB = kA + H_DIM / 2;
          v2f aA = *(const v2f*)(scur + m * SROW + kA);  // LDS, conflict-free
          v2f aB = *(const v2f*)(scur + m * SROW + kB);
          v2f bA = *(const v2f*)(wh + kA);
          v2f bB = *(const v2f*)(wh + kB);
          accA = wmma_f32_k4(aA, bA, accA);
          accB = wmma_f32_k4(aB, bB, accB);
        }
      }
      v8f sn;
#pragma unroll
      for (int r = 0; r < 8; ++r) sn[r] = tanhf(accA[r] + accB[r]);

      // publish s_new to the other buffer
      {
        float* snxt = &s_ls[cur ^ 1][0];
#pragma unroll
        for (int r = 0; r < 8; ++r) snxt[(8 * khi + r) * SROW + col] = sn[r];
      }
      __syncthreads();  // (1) s_new complete

      // ---- halt matvec: wave w handles row w, 8 MACs per lane ----
      {
        const float* sr = &s_ls[cur ^ 1][0] + wave * SROW + lane * 8;
        const float* wv = whalt_s + lane * 8;
        float ps = 0.0f;
#pragma unroll
        for (int j = 0; j < 8; ++j) ps += sr[j] * wv[j];
        partial[wave][lane] = ps;
      }
      __syncthreads();  // (2) partials ready

      // per-row halting bookkeeping (lanes 0-15 of wave 0); y-GEMM below does
      // not need p, so no barrier here -- the GEMM hides this scalar phase.
      if (tid < 16) {
        float hr = 0.0f;
#pragma unroll
        for (int j = 0; j < 32; ++j) hr += partial[tid][j];
        const float h      = 1.0f / (1.0f + expf(-(hr + bh)));
        const float hs     = hsum_s[tid];
        const float nsum   = hs + h;
        const bool  halted = (nsum >= 1.0f - 0.01f);
        const float run    = run_s[tid];
        const float p      = (halted ? (1.0f - hs) : h) * run;
        p_s[tid] = p;
        if (halted && run > 0.5f) rem_s[tid] = 1.0f - hs;
        nst_s[tid] += run;
        if (halted) run_s[tid] = 0.0f;
        hsum_s[tid] = nsum;
      }

      // ---- y_n = s_new @ W_out.T ; two K-split chains ----
      v8f yaA = v8_splat(0.0f), yaB = v8_splat(0.0f);
      {
        const float* snew = &s_ls[cur ^ 1][0];
        const float* wo   = W_out + (size_t)col * H_DIM;
#pragma unroll 8
        for (int k0 = 0; k0 < H_DIM / 2; k0 += 4) {
          const int kA = k0 + 2 * khi;
          const int kB = kA + H_DIM / 2;
          v2f aA = *(const v2f*)(snew + m * SROW + kA);
          v2f aB = *(const v2f*)(snew + m * SROW + kB);
          v2f bA = *(const v2f*)(wo + kA);
          v2f bB = *(const v2f*)(wo + kB);
          yaA = wmma_f32_k4(aA, bA, yaA);
          yaB = wmma_f32_k4(aB, bB, yaB);
        }
      }
      __syncthreads();  // (3) p_s now visible

      float pr[8];
#pragma unroll
      for (int r = 0; r < 8; ++r) pr[r] = p_s[8 * khi + r];
#pragma unroll
      for (int r = 0; r < 8; ++r) {
        sacc[r] += pr[r] * sn[r];
        yacc[r] += pr[r] * (yaA[r] + yaB[r] + bo);
      }

      cur ^= 1;
    }  // n loop

    __syncthreads();  // all reads of final s_new done before overwrite

    // s_prev for next t = s_acc (cur is back to its start-of-t value)
    {
      float* sdst = &s_ls[cur][0];
#pragma unroll
      for (int r = 0; r < 8; ++r) sdst[(8 * khi + r) * SROW + col] = sacc[r];
    }

    // outputs for timestep t
    {
      float* yb = y_out + ((size_t)t * B_DIM + b0) * O_DIM;
#pragma unroll
      for (int r = 0; r < 8; ++r)
        yb[(size_t)(8 * khi + r) * O_DIM + col] = yacc[r];
    }
    if (tid < 16) {
      p_out[(size_t)t * B_DIM + b0 + tid] = nst_s[tid] + rem_s[tid];
      n_out[(size_t)t * B_DIM + b0 + tid] = nst_s[tid];
      hsum_s[tid] = 0.0f; rem_s[tid] = 0.0f; nst_s[tid] = 0.0f; run_s[tid] = 1.0f;
    }
    __syncthreads();  // s_prev published, halt state reset, xs reusable
  }  // t loop
}

extern "C" void kernel_launch(void* const* d_in, const int* in_sizes, int n_in,
                              void* d_out, int out_size, void* d_ws, size_t ws_size,
                              hipStream_t stream) {
  (void)in_sizes; (void)n_in; (void)out_size; (void)d_ws; (void)ws_size;
  const float* x      = (const float*)d_in[0];
  const float* W_ih   = (const float*)d_in[1];
  const float* b_ih   = (const float*)d_in[2];
  const float* W_hh   = (const float*)d_in[3];
  const float* b_hh   = (const float*)d_in[4];
  const float* W_halt = (const float*)d_in[5];
  const float* b_halt = (const float*)d_in[6];
  const float* W_out  = (const float*)d_in[7];
  const float* b_out  = (const float*)d_in[8];

  float* out   = (float*)d_out;
  float* y_out = out;                                   // [S,B,O]
  float* p_out = out + (size_t)S_DIM * B_DIM * O_DIM;   // [S,B]
  float* n_out = p_out + (size_t)S_DIM * B_DIM;         // [S,B]

  dim3 grid(B_DIM / 16);   // 32 workgroups, each owns 16 batch rows for all t
  dim3 block(512);         // 16 wave32s, one 16-col N-tile per wave
  act_persistent_kernel<<<grid, block, 0, stream>>>(
      x, W_ih, b_ih, W_hh, b_hh, W_halt, b_halt, W_out, b_out,
      y_out, p_out, n_out);
}